// KnnConstraint_38457137169061
// MI455X (gfx1250) — compile-verified
//
#include <hip/hip_runtime.h>
#include <stdint.h>

// Problem constants (from reference setup_inputs / module constants)
#define NPTS 4096
#define NB   4
#define KNN  20
#define TEMP 0.1f

#define THREADS 128
#define BLOCKS_PER_BATCH (NPTS / THREADS)   // 32
#define NBLOCKS (NB * BLOCKS_PER_BATCH)     // 128

#ifndef __has_builtin
#define __has_builtin(x) 0
#endif

#if __has_builtin(__builtin_amdgcn_global_load_async_to_lds_b128)
#define HAVE_ASYNC_LDS 1
#else
#define HAVE_ASYNC_LDS 0
#endif

// The async-LDS builtins take int4 pointers in explicit address spaces:
//   void __builtin_amdgcn_global_load_async_to_lds_b128(v4i AS1*, v4i AS3*, imm, imm)
typedef int v4i __attribute__((ext_vector_type(4)));
typedef __attribute__((address_space(1))) v4i gv4i;  // global
typedef __attribute__((address_space(3))) v4i lv4i;  // LDS

__device__ __forceinline__ gv4i* gcast(const void* p) {
    // C-style casts: drop const, then generic->AS1 addrspacecast
    return (gv4i*)(void*)p;
}
__device__ __forceinline__ lv4i* lcast(void* p) {
    // generic->AS3 addrspacecast
    return (lv4i*)p;
}

__global__ __launch_bounds__(THREADS)
void knn_loss_kernel(const float* __restrict__ xyz,
                     const float* __restrict__ can,
                     const float* __restrict__ radius,
                     float* __restrict__ block_sums) {
    // LDS: current-frame batch points, canonical points, |p|^2 table, reduction pad
    __shared__ __align__(16) float s_xyz[NPTS * 3];   // 48 KB
    __shared__ __align__(16) float s_can[NPTS * 3];   // 48 KB
    __shared__ float s_sq[NPTS];                      // 16 KB
    __shared__ float s_red[THREADS];

    const int batch = blockIdx.x / BLOCKS_PER_BATCH;
    const int i     = (blockIdx.x % BLOCKS_PER_BATCH) * THREADS + (int)threadIdx.x;
    const float* gx = xyz + (size_t)batch * NPTS * 3;

    // ---- Stage 48KB + 48KB into LDS via CDNA5 async global->LDS path ----
#if HAVE_ASYNC_LDS
    {
        const char* g0 = (const char*)gx;
        const char* g1 = (const char*)can;
        char* l0 = (char*)s_xyz;
        char* l1 = (char*)s_can;
        // NPTS*3*4 = 49152 bytes = 3072 x 16B chunks; 3072 % 128 == 0 (no divergence)
        for (int k = (int)threadIdx.x * 16; k < NPTS * 3 * 4; k += THREADS * 16) {
            __builtin_amdgcn_global_load_async_to_lds_b128(gcast(g0 + k), lcast(l0 + k), 0, 0);
            __builtin_amdgcn_global_load_async_to_lds_b128(gcast(g1 + k), lcast(l1 + k), 0, 0);
        }
#if __has_builtin(__builtin_amdgcn_s_wait_asynccnt)
        __builtin_amdgcn_s_wait_asynccnt(0);
#else
        asm volatile("s_wait_asynccnt 0" ::: "memory");
#endif
    }
#else
    {
        const float4* g0 = (const float4*)gx;
        const float4* g1 = (const float4*)can;
        float4* l0 = (float4*)s_xyz;
        float4* l1 = (float4*)s_can;
        for (int k = (int)threadIdx.x; k < (NPTS * 3) / 4; k += THREADS) {
            l0[k] = g0[k];
            l1[k] = g1[k];
        }
    }
#endif
    __syncthreads();

    // Precompute |p|^2 (same fma ordering as the dot below so self-d2 == 0 exactly)
    for (int k = (int)threadIdx.x; k < NPTS; k += THREADS) {
        float x = s_xyz[3 * k], y = s_xyz[3 * k + 1], z = s_xyz[3 * k + 2];
        s_sq[k] = fmaf(z, z, fmaf(y, y, x * x));
    }
    __syncthreads();

    const float r  = *radius;
    const float r2 = r * r;

    const float xi = s_xyz[3 * i], yi = s_xyz[3 * i + 1], zi = s_xyz[3 * i + 2];
    const float sqi = s_sq[i];
    const float cxi = s_can[3 * i], cyi = s_can[3 * i + 1], czi = s_can[3 * i + 2];

    // Ordered ball-query scan with early exit: first KNN in-ball indices,
    // drop hit #1 (reference drops column 0 of top-K), stream terms for 2..KNN.
    int   c   = 0;
    float acc = 0.0f;
    for (int j = 0; j < NPTS; ++j) {
        float xj = s_xyz[3 * j], yj = s_xyz[3 * j + 1], zj = s_xyz[3 * j + 2];
        float dot = fmaf(zi, zj, fmaf(yi, yj, xi * xj));
        float d2  = fmaf(-2.0f, dot, sqi + s_sq[j]);   // GEMM form, matches reference
        if (d2 <= r2) {
            ++c;
            if (c >= 2) {
                // current-frame distance (direct form, like reference's safe_norm)
                float dx = xi - xj, dy = yi - yj, dz = zi - zj;
                float cd = sqrtf(fmaf(dz, dz, fmaf(dy, dy, dx * dx)));
                // canonical distance with the same neighbor index
                float ex = s_can[3 * j], ey = s_can[3 * j + 1], ez = s_can[3 * j + 2];
                float gx2 = cxi - ex, gy2 = cyi - ey, gz2 = czi - ez;
                float cnd = sqrtf(fmaf(gz2, gz2, fmaf(gy2, gy2, gx2 * gx2)));
                float w    = expf(-(cnd * cnd) * TEMP);
                float diff = cd - cnd;
                acc += sqrtf(fmaf(diff * diff, w, 1e-20f));
                if (c == KNN) break;
            }
        }
    }
    // Slots never filled contribute sqrt(1e-20) = 1e-10 each (weight masked to 0)
    acc += (float)(KNN - c) * 1e-10f;

    // Deterministic block reduction
    s_red[threadIdx.x] = acc;
    __syncthreads();
    for (int s = THREADS / 2; s > 0; s >>= 1) {
        if ((int)threadIdx.x < s) s_red[threadIdx.x] += s_red[threadIdx.x + s];
        __syncthreads();
    }
    if (threadIdx.x == 0) block_sums[blockIdx.x] = s_red[0];
}

__global__ __launch_bounds__(NBLOCKS)
void knn_finalize_kernel(const float* __restrict__ block_sums, float* __restrict__ out) {
    __shared__ double s_red[NBLOCKS];
    int t = (int)threadIdx.x;
    s_red[t] = (double)block_sums[t];
    __syncthreads();
    for (int s = NBLOCKS / 2; s > 0; s >>= 1) {
        if (t < s) s_red[t] += s_red[t + s];
        __syncthreads();
    }
    if (t == 0) {
        const double scale = 1.0 / ((double)NB * (double)NPTS * (double)(KNN - 1)); // 1/311296
        out[0] = (float)(s_red[0] * scale);
    }
}

extern "C" void kernel_launch(void* const* d_in, const int* in_sizes, int n_in,
                              void* d_out, int out_size, void* d_ws, size_t ws_size,
                              hipStream_t stream) {
    (void)in_sizes; (void)n_in; (void)out_size; (void)ws_size;
    const float* xyz    = (const float*)d_in[0];  // [4,4096,3] f32
    const float* can    = (const float*)d_in[1];  // [4096,3]   f32
    const float* radius = (const float*)d_in[2];  // scalar     f32
    float* out   = (float*)d_out;                 // scalar     f32
    float* bsums = (float*)d_ws;                  // NBLOCKS floats of scratch

    knn_loss_kernel<<<NBLOCKS, THREADS, 0, stream>>>(xyz, can, radius, bsums);
    knn_finalize_kernel<<<1, NBLOCKS, 0, stream>>>(bsums, out);
}